// LeakyIntegrator_57638461112614
// MI455X (gfx1250) — compile-verified
//
#include <hip/hip_runtime.h>
#include <stdint.h>

// ---------------------------------------------------------------------------
// LeakyIntegrator scan for MI455X (gfx1250, wave32).
//   delta_t = |stick[seq_t]| - |clear| ;  out_t = S_t - min(0, cummin(S_t))
// Monoid: (s1,m1) o (s2,m2) = (s1+s2, min(m1, s1+m2)), identity (0, +inf).
// One workgroup per row; async global->LDS staging, wave32 shuffle scans.
// ---------------------------------------------------------------------------

constexpr int L_LEN   = 131072;            // sequence length (reference-fixed)
constexpr int THREADS = 1024;              // 32 waves of 32 lanes
constexpr int NWAVES  = THREADS / 32;      // 32
constexpr int REGION  = L_LEN / NWAVES;    // 4096 elements per wave
constexpr int TILE    = 512;               // elements per wave tile (2 KiB)
constexpr int SEG     = 16;                // elements per lane per tile
constexpr int NT      = REGION / TILE;     // 8 tiles per wave

static_assert(TILE == 32 * SEG, "tile layout");
static_assert(REGION % TILE == 0, "region layout");
static_assert(NT >= 3, "async pipeline needs >= 3 tiles");

// ---- gfx1250 async-tensor path (inline asm; portable across toolchains) ----
__device__ __forceinline__ void async_load_b128_to_lds(uint32_t lds_addr,
                                                       const void* gaddr) {
  asm volatile("global_load_async_to_lds_b128 %0, %1, off"
               :: "v"(lds_addr), "v"((uint64_t)(uintptr_t)gaddr)
               : "memory");
}
template <int N>
__device__ __forceinline__ void wait_asynccnt() {
  asm volatile("s_wait_asynccnt %0" :: "i"(N) : "memory");
}
__device__ __forceinline__ void wait_dscnt0() {
  asm volatile("s_wait_dscnt 0" ::: "memory");
}

// select delta for seq value q in {0,1,2,3} (2 cndmask chains, no LDS table)
__device__ __forceinline__ float dsel(int q, float d0, float d1, float d2, float d3) {
  float lo = (q & 1) ? d1 : d0;
  float hi = (q & 1) ? d3 : d2;
  return (q & 2) ? hi : lo;
}

// wave32 inclusive scan of the (sum, min-prefix) monoid
__device__ __forceinline__ void wave_scan_incl(float& s, float& m, int lane) {
  #pragma unroll
  for (int d = 1; d < 32; d <<= 1) {
    float ps = __shfl_up(s, (unsigned)d, 32);
    float pm = __shfl_up(m, (unsigned)d, 32);
    if (lane >= d) { m = fminf(pm, ps + m); s += ps; }
  }
}

__global__ __launch_bounds__(THREADS, 1)
void LeakyIntegrator_57638461112614_kernel(const int* __restrict__ seq,
                                           const float* __restrict__ stick,
                                           const float* __restrict__ clear_rate,
                                           float* __restrict__ out) {
  // double-buffered coalesced staging: 32 waves * 2 * 2048 B = 128 KiB LDS
  __shared__ __align__(16) unsigned char stage[NWAVES][2][TILE * 4];
  __shared__ float wS[NWAVES], wM[NWAVES], wPS[NWAVES], wPM[NWAVES];

  const int tid  = threadIdx.x;
  const int wv   = tid >> 5;
  const int lane = tid & 31;

  const float ca = fabsf(clear_rate[0]);
  const float d0 = fabsf(stick[0]) - ca;
  const float d1 = fabsf(stick[1]) - ca;
  const float d2 = fabsf(stick[2]) - ca;
  const float d3 = fabsf(stick[3]) - ca;

  const int* wreg = seq + (size_t)blockIdx.x * L_LEN + wv * REGION;
  float*     wout = out + (size_t)blockIdx.x * L_LEN + wv * REGION;

  uint32_t ldsb[2];
  ldsb[0] = (uint32_t)(uintptr_t)&stage[wv][0][0];   // low 32 bits = LDS offset
  ldsb[1] = (uint32_t)(uintptr_t)&stage[wv][1][0];

  // issue one 2 KiB tile: 4 x 512 B fully-coalesced async b128 loads.
  // XOR-swizzle 16 B quads so the strided per-lane ds_load_b128 reads
  // (lane stride 64 B) spread across banks without padding.
  auto issue = [&](int t) {
    uint32_t base = ldsb[t & 1];
    const unsigned char* g =
        (const unsigned char*)(wreg + t * TILE) + lane * 16;
    #pragma unroll
    for (int j = 0; j < 4; ++j) {
      int seg = 8 * j + (lane >> 2);                   // segment = owning lane
      int kk  = (lane & 3) ^ ((seg >> 2) & 3);         // swizzled quad slot
      async_load_b128_to_lds(base + (uint32_t)(seg * 64 + kk * 16), g + j * 512);
    }
  };
  // read quad k (elements 4k..4k+3) of this lane's segment for tile t
  auto read4 = [&](int t, int k) -> int4 {
    int off = lane * 64 + ((k ^ ((lane >> 2) & 3)) * 16);
    return *(const int4*)&stage[wv][t & 1][off];
  };

  const float INF = __builtin_inff();

  // ---------------- phase 1: per-wave (sum, min-prefix) totals --------------
  issue(0); issue(1);
  float accS = 0.f, accM = INF;
  #pragma unroll 1
  for (int t = 0; t < NT; ++t) {
    if (t < NT - 1) wait_asynccnt<4>(); else wait_asynccnt<0>();
    float s = 0.f, m = INF;
    #pragma unroll
    for (int k = 0; k < 4; ++k) {
      int4 v = read4(t, k);
      float x;
      x = dsel(v.x, d0, d1, d2, d3); s += x; m = fminf(m, s);
      x = dsel(v.y, d0, d1, d2, d3); s += x; m = fminf(m, s);
      x = dsel(v.z, d0, d1, d2, d3); s += x; m = fminf(m, s);
      x = dsel(v.w, d0, d1, d2, d3); s += x; m = fminf(m, s);
    }
    wave_scan_incl(s, m, lane);
    float ts = __shfl(s, 31, 32);
    float tm = __shfl(m, 31, 32);
    accM = fminf(accM, accS + tm);
    accS += ts;
    if (t + 2 < NT) { wait_dscnt0(); issue(t + 2); }   // reuse drained buffer
  }

  if (lane == 0) { wS[wv] = accS; wM[wv] = accM; }
  __syncthreads();
  if (wv == 0) {
    float s2 = (lane < NWAVES) ? wS[lane] : 0.f;
    float m2 = (lane < NWAVES) ? wM[lane] : INF;
    wave_scan_incl(s2, m2, lane);
    float es = __shfl_up(s2, 1u, 32);
    float em = __shfl_up(m2, 1u, 32);
    if (lane == 0) { es = 0.f; em = INF; }
    if (lane < NWAVES) { wPS[lane] = es; wPM[lane] = em; }
  }
  __syncthreads();

  float runS = wPS[wv];
  float runM = fminf(wPM[wv], 0.f);      // fold in global seed S_0 = 0

  // ---------------- phase 2: emit outputs (seq re-read hits L2) -------------
  wait_dscnt0();
  issue(0); issue(1);
  #pragma unroll 1
  for (int t = 0; t < NT; ++t) {
    if (t < NT - 1) wait_asynccnt<4>(); else wait_asynccnt<0>();
    float dl[SEG];
    float s = 0.f, m = INF;
    #pragma unroll
    for (int k = 0; k < 4; ++k) {
      int4 v = read4(t, k);
      float x;
      x = dsel(v.x, d0, d1, d2, d3); dl[4*k+0] = x; s += x; m = fminf(m, s);
      x = dsel(v.y, d0, d1, d2, d3); dl[4*k+1] = x; s += x; m = fminf(m, s);
      x = dsel(v.z, d0, d1, d2, d3); dl[4*k+2] = x; s += x; m = fminf(m, s);
      x = dsel(v.w, d0, d1, d2, d3); dl[4*k+3] = x; s += x; m = fminf(m, s);
    }
    float is = s, im = m;
    wave_scan_incl(is, im, lane);
    float es = __shfl_up(is, 1u, 32);
    float em = __shfl_up(im, 1u, 32);
    if (lane == 0) { es = 0.f; em = INF; }
    float r  = runS + es;                       // S before this segment
    float cm = fminf(runM, runS + em);          // min(0, cummin) so far
    float* o = wout + t * TILE + lane * SEG;
    #pragma unroll
    for (int k = 0; k < 4; ++k) {
      float4 q;
      r += dl[4*k+0]; cm = fminf(cm, r); q.x = r - cm;
      r += dl[4*k+1]; cm = fminf(cm, r); q.y = r - cm;
      r += dl[4*k+2]; cm = fminf(cm, r); q.z = r - cm;
      r += dl[4*k+3]; cm = fminf(cm, r); q.w = r - cm;
      *(float4*)(o + 4 * k) = q;
    }
    float ts = __shfl(is, 31, 32);
    float tm = __shfl(im, 31, 32);
    runM = fminf(runM, runS + tm);
    runS += ts;
    if (t + 2 < NT) { wait_dscnt0(); issue(t + 2); }
  }
}

extern "C" void kernel_launch(void* const* d_in, const int* in_sizes, int n_in,
                              void* d_out, int out_size, void* d_ws, size_t ws_size,
                              hipStream_t stream) {
  (void)n_in; (void)out_size; (void)d_ws; (void)ws_size;
  const int*   seq = (const int*)d_in[0];
  const float* st  = (const float*)d_in[1];
  const float* cl  = (const float*)d_in[2];
  float*       out = (float*)d_out;
  const int B = in_sizes[0] / L_LEN;    // 128 rows
  LeakyIntegrator_57638461112614_kernel<<<dim3(B), dim3(THREADS), 0, stream>>>(
      seq, st, cl, out);
}